// S6DReal_40123584479404
// MI455X (gfx1250) — compile-verified
//
#include <hip/hip_runtime.h>
#include <math.h>

#define L_SEQ 4096
#define C_DIM 128
#define S_DIM 64
#define K_DIM 128
#define NCH   64          // number of scan chunks
#define CHLEN 64          // L_SEQ / NCH

typedef float v2f __attribute__((ext_vector_type(2)));
typedef float v8f __attribute__((ext_vector_type(8)));

__device__ __forceinline__ float softplus_f(float z) {
    // softplus(z) = log(1+exp(z)); guard large z to avoid inf in exp
    return (z > 20.0f) ? z : logf(1.0f + __expf(z));
}

// -----------------------------------------------------------------------------
// Kernel 1: fused projection GEMM.  Computes the 4096x256 product of
// x (4096x128) with [B_kernel | C_kernel | dt_kernel] using f32 WMMA
// (V_WMMA_F32_16X16X4_F32), one wave per 16x16 output tile, K-loop of 32.
// Epilogue fuses bias, the "1 +" for B, and softplus for dt.
// -----------------------------------------------------------------------------
__global__ __launch_bounds__(256)
void proj_gemm_kernel(const float* __restrict__ x,
                      const float* __restrict__ Bk, const float* __restrict__ Bb,
                      const float* __restrict__ Ck, const float* __restrict__ Cb,
                      const float* __restrict__ Dk, const float* __restrict__ Db,
                      float* __restrict__ Bws, float* __restrict__ Cws,
                      float* __restrict__ dtws) {
    const int tid  = threadIdx.x;
    const int lane = tid & 31;
    const int wave = tid >> 5;
    const int w      = blockIdx.x * 8 + wave;   // 4096 tiles total
    const int tile_m = w >> 4;                  // 0..255
    const int tile_n = w & 15;                  // 0..15
    const int m0 = tile_m * 16;
    const int n0 = tile_n * 16;

    const int half = lane >> 4;                 // 0 or 1 (lanes 16-31)
    const int mrow = lane & 15;
    const int n    = n0 + mrow;                 // column this lane handles

    // Section select is wave-uniform (sections are multiples of 64 cols).
    const float* wcol;  int ld;
    if (n0 < 64)        { wcol = Bk + n;         ld = S_DIM; }
    else if (n0 < 128)  { wcol = Ck + (n - 64);  ld = S_DIM; }
    else                { wcol = Dk + (n - 128); ld = C_DIM; }

    const float* arow = x + (size_t)(m0 + mrow) * K_DIM;

    v8f acc = {};
    for (int k0 = 0; k0 < K_DIM; k0 += 4) {
        const int ka = k0 + 2 * half;
        v2f a, b;
        // A 16x4 f32 layout: lanes 0-15 K=0/1, lanes 16-31 K=2/3, M=lane%16
        a.x = arow[ka];
        a.y = arow[ka + 1];
        // B 4x16 f32 layout: row K striped across lanes (N=lane%16)
        b.x = wcol[(size_t)ka * ld];
        b.y = wcol[(size_t)(ka + 1) * ld];
        acc = __builtin_amdgcn_wmma_f32_16x16x4_f32(false, a, false, b,
                                                    (short)0, acc, false, false);
    }

    // C/D layout: VGPR r -> M = r (lanes 0-15) or r+8 (lanes 16-31), N = lane%16
    if (n0 < 64) {
        const float bias = Bb[n];
        #pragma unroll
        for (int r = 0; r < 8; ++r) {
            const int row = m0 + r + 8 * half;
            Bws[(size_t)row * S_DIM + n] = 1.0f + acc[r] + bias;
        }
    } else if (n0 < 128) {
        const float bias = Cb[n - 64];
        #pragma unroll
        for (int r = 0; r < 8; ++r) {
            const int row = m0 + r + 8 * half;
            Cws[(size_t)row * S_DIM + (n - 64)] = acc[r] + bias;
        }
    } else {
        const float bias = Db[n - 128];
        #pragma unroll
        for (int r = 0; r < 8; ++r) {
            const int row = m0 + r + 8 * half;
            const float z = (1.0f / 4096.0f) + acc[r] + bias;
            dtws[(size_t)row * C_DIM + (n - 128)] = softplus_f(z);
        }
    }
}

// -----------------------------------------------------------------------------
// Kernel 2: pass 1 of chunked scan. One wave32 block per (chunk, channel);
// lane owns the adjacent pair s = 2*lane, 2*lane+1 so all per-s state loads
// are 8-byte vector loads (global_load_b64).  Broadcast operands dt[l,c] and
// x[l,c] are staged in LDS once per chunk and re-read via ds_load broadcasts.
// -----------------------------------------------------------------------------
__global__ __launch_bounds__(32)
void scan_pass1_kernel(const float* __restrict__ x,
                       const float* __restrict__ A_log,
                       const float* __restrict__ Bws,
                       const float* __restrict__ dtws,
                       float* __restrict__ chunkA,
                       float* __restrict__ chunkH) {
    __shared__ float sdt[CHLEN];
    __shared__ float sx[CHLEN];

    const int lane = threadIdx.x;
    const int c    = blockIdx.x & (C_DIM - 1);
    const int ch   = blockIdx.x >> 7;
    const int l0   = ch * CHLEN;

    // Stage the broadcast column (dt, x) for this chunk into LDS.
    sdt[lane]      = dtws[(l0 + lane) * C_DIM + c];
    sdt[lane + 32] = dtws[(l0 + lane + 32) * C_DIM + c];
    sx[lane]       = x[(l0 + lane) * C_DIM + c];
    sx[lane + 32]  = x[(l0 + lane + 32) * C_DIM + c];
    __syncthreads();          // single-wave workgroup: degrades to S_NOP

    const v2f Alg = *(const v2f*)&A_log[c * S_DIM + 2 * lane];
    const float A0  = -__expf(Alg.x);
    const float A1  = -__expf(Alg.y);
    const float iA0 = 1.0f / A0, iA1 = 1.0f / A1;

    float h0 = 0.f, h1 = 0.f, p0 = 1.f, p1 = 1.f;
    #pragma unroll 4
    for (int t = 0; t < CHLEN; ++t) {
        const int l = l0 + t;
        const float dt = sdt[t];
        const float xv = sx[t];
        const v2f bv = *(const v2f*)&Bws[l * S_DIM + 2 * lane];
        const float e0 = __expf(A0 * dt);
        const float e1 = __expf(A1 * dt);
        const float u0 = (e0 - 1.0f) * (bv.x * iA0) * xv;
        const float u1 = (e1 - 1.0f) * (bv.y * iA1) * xv;
        h0 = fmaf(e0, h0, u0);
        h1 = fmaf(e1, h1, u1);
        p0 *= e0;
        p1 *= e1;
    }
    const size_t base = ((size_t)ch * C_DIM + c) * S_DIM + 2 * lane;
    v2f pv; pv.x = p0; pv.y = p1;
    v2f hv; hv.x = h0; hv.y = h1;
    *(v2f*)&chunkA[base] = pv;
    *(v2f*)&chunkH[base] = hv;
}

// -----------------------------------------------------------------------------
// Kernel 3: middle scan over the 64 chunk aggregates per chain -> exclusive
// carry-in h for every chunk. One wave32 block per channel, 2 s per lane,
// all accesses 8-byte vector ops.
// -----------------------------------------------------------------------------
__global__ __launch_bounds__(32)
void mid_scan_kernel(const float* __restrict__ chunkA,
                     const float* __restrict__ chunkH,
                     float* __restrict__ carry) {
    const int lane = threadIdx.x;
    const int c    = blockIdx.x;
    float c0 = 0.f, c1 = 0.f;
    for (int ch = 0; ch < NCH; ++ch) {
        const size_t base = ((size_t)ch * C_DIM + c) * S_DIM + 2 * lane;
        v2f cv; cv.x = c0; cv.y = c1;
        *(v2f*)&carry[base] = cv;
        const v2f av = *(const v2f*)&chunkA[base];
        const v2f hv = *(const v2f*)&chunkH[base];
        c0 = fmaf(av.x, c0, hv.x);
        c1 = fmaf(av.y, c1, hv.y);
    }
}

// -----------------------------------------------------------------------------
// Kernel 4: pass 2. Re-runs each chunk with the correct carry-in and folds
// y[l,c] = sum_s Cm[l,s]*h[l,c,s] via a wave32 shuffle reduction (64 s values
// = 2 adjacent per lane, so the reduction stays inside one wave; no LDS).
// -----------------------------------------------------------------------------
__global__ __launch_bounds__(32)
void scan_pass2_kernel(const float* __restrict__ x,
                       const float* __restrict__ A_log,
                       const float* __restrict__ Bws,
                       const float* __restrict__ Cws,
                       const float* __restrict__ dtws,
                       const float* __restrict__ carry,
                       float* __restrict__ y) {
    __shared__ float sdt[CHLEN];
    __shared__ float sx[CHLEN];

    const int lane = threadIdx.x;
    const int c    = blockIdx.x & (C_DIM - 1);
    const int ch   = blockIdx.x >> 7;
    const int l0   = ch * CHLEN;

    sdt[lane]      = dtws[(l0 + lane) * C_DIM + c];
    sdt[lane + 32] = dtws[(l0 + lane + 32) * C_DIM + c];
    sx[lane]       = x[(l0 + lane) * C_DIM + c];
    sx[lane + 32]  = x[(l0 + lane + 32) * C_DIM + c];
    __syncthreads();

    const v2f Alg = *(const v2f*)&A_log[c * S_DIM + 2 * lane];
    const float A0  = -__expf(Alg.x);
    const float A1  = -__expf(Alg.y);
    const float iA0 = 1.0f / A0, iA1 = 1.0f / A1;

    const size_t cbase = ((size_t)ch * C_DIM + c) * S_DIM + 2 * lane;
    const v2f h00 = *(const v2f*)&carry[cbase];
    float h0 = h00.x, h1 = h00.y;

    #pragma unroll 2
    for (int t = 0; t < CHLEN; ++t) {
        const int l = l0 + t;
        const float dt = sdt[t];
        const float xv = sx[t];
        const v2f bv = *(const v2f*)&Bws[l * S_DIM + 2 * lane];
        const v2f cv = *(const v2f*)&Cws[l * S_DIM + 2 * lane];
        const float e0 = __expf(A0 * dt);
        const float e1 = __expf(A1 * dt);
        const float u0 = (e0 - 1.0f) * (bv.x * iA0) * xv;
        const float u1 = (e1 - 1.0f) * (bv.y * iA1) * xv;
        h0 = fmaf(e0, h0, u0);
        h1 = fmaf(e1, h1, u1);

        float p = cv.x * h0 + cv.y * h1;
        #pragma unroll
        for (int off = 16; off > 0; off >>= 1)
            p += __shfl_xor(p, off, 32);
        if (lane == 0) y[l * C_DIM + c] = p;
    }
}

// -----------------------------------------------------------------------------
extern "C" void kernel_launch(void* const* d_in, const int* in_sizes, int n_in,
                              void* d_out, int out_size, void* d_ws, size_t ws_size,
                              hipStream_t stream) {
    (void)in_sizes; (void)n_in; (void)out_size; (void)ws_size;
    const float* x  = (const float*)d_in[0];   // (L, C)
    const float* Al = (const float*)d_in[1];   // (C, S)
    const float* Bk = (const float*)d_in[2];   // (C, S)
    const float* Bb = (const float*)d_in[3];   // (S,)
    const float* Ck = (const float*)d_in[4];   // (C, S)
    const float* Cb = (const float*)d_in[5];   // (S,)
    const float* Dk = (const float*)d_in[6];   // (C, C)
    const float* Db = (const float*)d_in[7];   // (C,)
    float* y = (float*)d_out;                  // (L, C)

    float* ws    = (float*)d_ws;
    float* Bws   = ws;                                      // L*S
    float* Cws   = Bws  + (size_t)L_SEQ * S_DIM;            // L*S
    float* dtws  = Cws  + (size_t)L_SEQ * S_DIM;            // L*C
    float* chA   = dtws + (size_t)L_SEQ * C_DIM;            // NCH*C*S
    float* chH   = chA  + (size_t)NCH * C_DIM * S_DIM;      // NCH*C*S
    float* carry = chH  + (size_t)NCH * C_DIM * S_DIM;      // NCH*C*S

    // 4096x256 projection: 4096 WMMA tiles, 8 waves/block -> 512 blocks.
    proj_gemm_kernel<<<512, 256, 0, stream>>>(x, Bk, Bb, Ck, Cb, Dk, Db,
                                              Bws, Cws, dtws);
    // Chunked linear-recurrence scan.
    scan_pass1_kernel<<<NCH * C_DIM, 32, 0, stream>>>(x, Al, Bws, dtws, chA, chH);
    mid_scan_kernel<<<C_DIM, 32, 0, stream>>>(chA, chH, carry);
    scan_pass2_kernel<<<NCH * C_DIM, 32, 0, stream>>>(x, Al, Bws, Cws, dtws,
                                                      carry, y);
}